// MotionDecoder_37615323578937
// MI455X (gfx1250) — compile-verified
//
#include <hip/hip_runtime.h>
#include <math.h>

#define DM 256
#define LAT 512
#define NB 32
#define NJ 143
#define NF 120
#define FEAT 13

typedef __attribute__((ext_vector_type(2))) float v2f;
typedef __attribute__((ext_vector_type(8))) float v8f;

// ---------------- workspace layout (floats) ----------------
#define OFF_BASE 0        // 32*256
#define OFF_A    8192     // 32*256   (base*gamma)@W1
#define OFF_B    16384    // 143*256  (joint*gamma)@W1
#define OFF_C    52992    // 120*256  (frame*gamma)@W1
#define OFF_MB   83712    // 32 mean(base)
#define OFF_QB   83744    // 32 mean(base^2)
#define OFF_MJ   83776    // 143
#define OFF_QJ   83920    // 143
#define OFF_MF   84064    // 120
#define OFF_QF   84184    // 120
#define OFF_BJ   84304    // 32*143  mean(base_b * joint_j)
#define OFF_BF   88880    // 32*120
#define OFF_JF   92720    // 143*120
#define OFF_GW   109880   // 256  gamma@W1
#define OFF_BW   110136   // 256  beta@W1 + b1

// Branch-free exact-gelu: 0.5*x*(1+erf(x/sqrt(2))), erf via A&S 7.1.26 (|err|<1.5e-7).
// Straight-line: v_rcp + v_exp + ~12 FMAs, keeps EXEC all-1s (WMMA requirement).
__device__ __forceinline__ float gelu_exact(float x) {
  float s = x * 0.70710678118654752f;
  float a = __builtin_fabsf(s);
  float t = __builtin_amdgcn_rcpf(__builtin_fmaf(0.3275911f, a, 1.0f));
  float p = __builtin_fmaf(t, 1.061405429f, -1.453152027f);
  p = __builtin_fmaf(t, p, 1.421413741f);
  p = __builtin_fmaf(t, p, -0.284496736f);
  p = __builtin_fmaf(t, p, 0.254829592f);
  p = p * t;
  float e = __builtin_amdgcn_exp2f(-a * a * 1.4426950408889634f);  // exp(-s^2)
  float erf_a = __builtin_fmaf(-p, e, 1.0f);
  float erf_s = __builtin_copysignf(erf_a, s);
  float hx = 0.5f * x;
  return __builtin_fmaf(hx, erf_s, hx);
}

// base = z @ W_latent + b_latent   (32 x 256)
__global__ __launch_bounds__(256) void k_base(const float* __restrict__ z,
                                              const float* __restrict__ Wl,
                                              const float* __restrict__ bl,
                                              float* __restrict__ base) {
  int b = blockIdx.x, d = threadIdx.x;
  float acc = bl[d];
  const float* zr = z + b * LAT;
  for (int c = 0; c < LAT; ++c) acc += zr[c] * Wl[c * DM + d];
  base[b * DM + d] = acc;
}

// per-row: mean, mean-of-squares, and G = (S ⊙ gamma) @ W1
__global__ __launch_bounds__(256) void k_rowtx(const float* __restrict__ S,
                                               const float* __restrict__ gamma,
                                               const float* __restrict__ W1,
                                               float* __restrict__ G,
                                               float* __restrict__ meanO,
                                               float* __restrict__ msqO) {
  int i = blockIdx.x, d = threadIdx.x;
  __shared__ float sg[DM];
  __shared__ float red[DM];
  float s = S[i * DM + d];
  sg[d] = s * gamma[d];
  red[d] = s;
  __syncthreads();
  for (int o = 128; o > 0; o >>= 1) { if (d < o) red[d] += red[d + o]; __syncthreads(); }
  float mean = red[0] * (1.0f / DM);
  __syncthreads();
  red[d] = s * s;
  __syncthreads();
  for (int o = 128; o > 0; o >>= 1) { if (d < o) red[d] += red[d + o]; __syncthreads(); }
  float msq = red[0] * (1.0f / DM);
  if (d == 0) { meanO[i] = mean; msqO[i] = msq; }
  float acc = 0.f;
  for (int c = 0; c < DM; ++c) acc += sg[c] * W1[c * DM + d];
  G[i * DM + d] = acc;
}

// out[i*ny+j] = mean_d( X[i,d]*Y[j,d] )
__global__ __launch_bounds__(256) void k_cross(const float* __restrict__ X,
                                               const float* __restrict__ Y,
                                               int ny, float* __restrict__ out) {
  int p = blockIdx.x, i = p / ny, j = p % ny, d = threadIdx.x;
  __shared__ float red[DM];
  red[d] = X[i * DM + d] * Y[j * DM + d];
  __syncthreads();
  for (int o = 128; o > 0; o >>= 1) { if (d < o) red[d] += red[d + o]; __syncthreads(); }
  if (d == 0) out[p] = red[0] * (1.0f / DM);
}

// gW1[d] = gamma@W1 ; bw[d] = beta@W1 + b1
__global__ __launch_bounds__(256) void k_vecs(const float* __restrict__ gamma,
                                              const float* __restrict__ beta,
                                              const float* __restrict__ W1,
                                              const float* __restrict__ b1,
                                              float* __restrict__ gW1,
                                              float* __restrict__ bw) {
  int d = threadIdx.x;
  float g = 0.f, bb = 0.f;
  for (int c = 0; c < DM; ++c) { float w = W1[c * DM + d]; g += gamma[c] * w; bb += beta[c] * w; }
  gW1[d] = g;
  bw[d] = bb + b1[d];
}

// Main fused kernel: one block per (b,j); 8 waves, each owns a 16-row f tile.
// pre[f,d] = inv*(AB[d] + C[f,d] - mu*gW1[d]) + bw[d]; h=gelu(pre); out = h @ W2pad via WMMA f32 16x16x4.
__global__ __launch_bounds__(256) void k_main(
    const float* __restrict__ A, const float* __restrict__ B, const float* __restrict__ C,
    const float* __restrict__ mB, const float* __restrict__ qB,
    const float* __restrict__ mJ, const float* __restrict__ qJ,
    const float* __restrict__ mF, const float* __restrict__ qF,
    const float* __restrict__ BJ, const float* __restrict__ BF, const float* __restrict__ JF,
    const float* __restrict__ gW1, const float* __restrict__ bw,
    const float* __restrict__ W2, const float* __restrict__ b2,
    float* __restrict__ out) {
  int b = blockIdx.x / NJ, j = blockIdx.x % NJ;
  int tid = threadIdx.x, lane = tid & 31, wave = tid >> 5;

  // W2 zero-padded to 256x16, pair-packed: pair (d,d+1) of column n at float2 index (d/2)*16+n
  __shared__ __align__(8) float s_w2[128 * 16 * 2];
  __shared__ __align__(8) float s_ab[DM];  // A_b + B_j
  __shared__ __align__(8) float s_g[DM];   // gamma@W1
  __shared__ __align__(8) float s_cc[DM];  // beta@W1 + b1
  {
    int d = tid;
    s_ab[d] = A[b * DM + d] + B[j * DM + d];
    s_g[d] = gW1[d];
    s_cc[d] = bw[d];
    for (int n = 0; n < 16; ++n) {
      float w = (n < FEAT) ? W2[d * FEAT + n] : 0.f;
      s_w2[((d >> 1) * 16 + n) * 2 + (d & 1)] = w;
    }
  }
  __syncthreads();

  int hi = lane >> 4;      // selects K pair {k0,k0+1} vs {k0+2,k0+3}
  int mrow = lane & 15;    // A-matrix row within tile / D-matrix column n
  int f0 = wave * 16;
  int f = f0 + mrow;
  int fc = f < NF ? f : NF - 1;  // clamp padded rows (stores are guarded)

  float mu = mB[b] + mJ[j] + mF[fc];
  float e2 = qB[b] + qJ[j] + qF[fc] + 2.f * (BJ[b * NJ + j] + BF[b * NF + fc] + JF[j * NF + fc]);
  float inv = rsqrtf(e2 - mu * mu + 1e-5f);

  const float2* crow = (const float2*)(C + fc * DM);
  const float2* ab2 = (const float2*)s_ab;
  const float2* g2 = (const float2*)s_g;
  const float2* cc2 = (const float2*)s_cc;
  const float2* w2p = (const float2*)s_w2;
  int n = mrow;

  v8f acc = {};
#pragma unroll 4
  for (int k0 = 0; k0 < DM; k0 += 4) {
    int d2 = (k0 >> 1) + hi;  // float2-pair index: d_base = k0 + 2*hi
    float2 cf = crow[d2];
    float2 ab = ab2[d2];
    float2 gg = g2[d2];
    float2 cc = cc2[d2];
    float px = inv * (ab.x + cf.x - mu * gg.x) + cc.x;
    float py = inv * (ab.y + cf.y - mu * gg.y) + cc.y;
    float hx = gelu_exact(px);
    float hy = gelu_exact(py);
    float2 bf = w2p[d2 * 16 + n];
    v2f av; av.x = hx; av.y = hy;
    v2f bv; bv.x = bf.x; bv.y = bf.y;
    acc = __builtin_amdgcn_wmma_f32_16x16x4_f32(false, av, false, bv, (short)0, acc, false, false);
  }

  // D layout: VGPR r -> M = r + 8*hi (f row), N = mrow (feature). Output is (b,j,feat,f).
  float b2n = (n < FEAT) ? b2[n] : 0.f;
  long obase = ((long)(b * NJ + j) * FEAT + n) * NF;
  if (n < FEAT) {
#pragma unroll
    for (int r = 0; r < 8; ++r) {
      int fo = f0 + r + hi * 8;
      if (fo < NF) out[obase + fo] = acc[r] + b2n;
    }
  }
}

extern "C" void kernel_launch(void* const* d_in, const int* in_sizes, int n_in,
                              void* d_out, int out_size, void* d_ws, size_t ws_size,
                              hipStream_t stream) {
  const float* z = (const float*)d_in[0];
  const float* W_latent = (const float*)d_in[1];
  const float* b_latent = (const float*)d_in[2];
  const float* joint_emb = (const float*)d_in[3];
  const float* frame_emb = (const float*)d_in[4];
  const float* ln_gamma = (const float*)d_in[5];
  const float* ln_beta = (const float*)d_in[6];
  const float* W1 = (const float*)d_in[7];
  const float* b1 = (const float*)d_in[8];
  const float* W2 = (const float*)d_in[9];
  const float* b2 = (const float*)d_in[10];
  float* ws = (float*)d_ws;
  float* out = (float*)d_out;

  k_base<<<NB, 256, 0, stream>>>(z, W_latent, b_latent, ws + OFF_BASE);
  k_rowtx<<<NB, 256, 0, stream>>>(ws + OFF_BASE, ln_gamma, W1, ws + OFF_A, ws + OFF_MB, ws + OFF_QB);
  k_rowtx<<<NJ, 256, 0, stream>>>(joint_emb, ln_gamma, W1, ws + OFF_B, ws + OFF_MJ, ws + OFF_QJ);
  k_rowtx<<<NF, 256, 0, stream>>>(frame_emb, ln_gamma, W1, ws + OFF_C, ws + OFF_MF, ws + OFF_QF);
  k_cross<<<NB * NJ, 256, 0, stream>>>(ws + OFF_BASE, joint_emb, NJ, ws + OFF_BJ);
  k_cross<<<NB * NF, 256, 0, stream>>>(ws + OFF_BASE, frame_emb, NF, ws + OFF_BF);
  k_cross<<<NJ * NF, 256, 0, stream>>>(joint_emb, frame_emb, NF, ws + OFF_JF);
  k_vecs<<<1, 256, 0, stream>>>(ln_gamma, ln_beta, W1, b1, ws + OFF_GW, ws + OFF_BW);

  k_main<<<NB * NJ, 256, 0, stream>>>(
      ws + OFF_A, ws + OFF_B, ws + OFF_C,
      ws + OFF_MB, ws + OFF_QB, ws + OFF_MJ, ws + OFF_QJ, ws + OFF_MF, ws + OFF_QF,
      ws + OFF_BJ, ws + OFF_BF, ws + OFF_JF,
      ws + OFF_GW, ws + OFF_BW, W2, b2, out);
}